// RuleNetwork_72421738545353
// MI455X (gfx1250) — compile-verified
//
#include <hip/hip_runtime.h>
#include <hip/hip_bf16.h>

// Sizes (hardcoded from reference)
#define B_    8192
#define NV_   2
#define CV_   1024
#define NR_   512
#define CR_   512
#define CM1_  32
#define CM2_  16
#define ROWS_ (B_ * NV_)   // 16384 rows of hidden, row = b*2 + n
#define NTILES_ 5          // [Wk1 lo | Wk1 hi | Wq2 | Wk2[0] | Wk2[1]]

typedef float v2f __attribute__((ext_vector_type(2)));
typedef float v4f __attribute__((ext_vector_type(4)));
typedef float v8f __attribute__((ext_vector_type(8)));

static __device__ __forceinline__ v8f zero8() {
  v8f z;
#pragma unroll
  for (int i = 0; i < 8; ++i) z[i] = 0.0f;
  return z;
}
static __device__ __forceinline__ v2f lo2(v4f a) { v2f r; r.x = a.x; r.y = a.y; return r; }
static __device__ __forceinline__ v2f hi2(v4f a) { v2f r; r.x = a.z; r.y = a.w; return r; }

// D = A(16x4 f32) * B(4x16 f32) + C(16x16 f32), exact fp32 WMMA
static __device__ __forceinline__ v8f wmma_f32(v2f a, v2f b, v8f c) {
  return __builtin_amdgcn_wmma_f32_16x16x4_f32(
      /*neg_a=*/false, a, /*neg_b=*/false, b,
      /*c_mod=*/(short)0, c, /*reuse_a=*/false, /*reuse_b=*/false);
}

// ---------------------------------------------------------------------------
// Kernel A1: r1 = rule_embeds @ Wq1 + bq1   (512 x 32) — tiny, plain dot.
// ---------------------------------------------------------------------------
__global__ __launch_bounds__(256) void r1_kernel(
    const float* __restrict__ re, const float* __restrict__ Wq1,
    const float* __restrict__ bq1, float* __restrict__ r1w) {
  int t = blockIdx.x * 256 + threadIdx.x;        // 0 .. 16383
  int r = t >> 5, m = t & 31;
  const float* rr = re + r * CR_;
  float acc = bq1[m];
#pragma unroll 8
  for (int k = 0; k < CR_; ++k) acc += rr[k] * Wq1[k * CM1_ + m];
  r1w[r * CM1_ + m] = acc;
}

// ---------------------------------------------------------------------------
// Kernel A2: pack weights transposed: WT[(t*16 + c)*1024 + row] so the
// phase-1 B operand is one b128 load per tile per 8-column chunk-pair.
//   t=0: Wk1[:, c]   t=1: Wk1[:, 16+c]   t=2: Wq2[:, c]
//   t=3: Wk2[0][:, c]   t=4: Wk2[1][:, c]
// ---------------------------------------------------------------------------
__global__ __launch_bounds__(256) void wt_kernel(
    const float* __restrict__ Wk1, const float* __restrict__ Wq2,
    const float* __restrict__ Wk2, float* __restrict__ WT) {
  int id = blockIdx.x * 256 + threadIdx.x;   // 0 .. 81919
  int tc = id >> 10;                         // t*16 + c
  int row = id & 1023;
  int t = tc >> 4, c = tc & 15;
  float v;
  if      (t == 0) v = Wk1[row * 32 + c];
  else if (t == 1) v = Wk1[row * 32 + 16 + c];
  else if (t == 2) v = Wq2[row * 16 + c];
  else if (t == 3) v = Wk2[row * 16 + c];
  else             v = Wk2[CV_ * CM2_ + row * 16 + c];
  WT[id] = v;
}

// ---------------------------------------------------------------------------
// Kernel B: fused projections + att1 max-reduce. One wave = one 16-row tile.
// K processed in pairs of chunks (8 cols) with permuted K-mapping so both
// A and B come from single b128 loads: chunk0 <- (x,y), chunk1 <- (z,w).
// ---------------------------------------------------------------------------
__global__ __launch_bounds__(256) void proj_att1_kernel(
    const float* __restrict__ hidden, const float* __restrict__ WT,
    const float* __restrict__ bk1, const float* __restrict__ bq2,
    const float* __restrict__ bk2, const float* __restrict__ r1w,
    float* __restrict__ p2o, float* __restrict__ k2o,
    float* __restrict__ bestvo, int* __restrict__ bestio) {
  __shared__ float lds_k1[8][16 * 32];   // per-wave k1 tile (2KB each)
  __shared__ float lds_rv[8][16 * 16];   // per-wave reduce values
  __shared__ int   lds_ri[8][16 * 16];   // per-wave reduce indices

  const int tid  = threadIdx.x;
  const int wave = tid >> 5;
  const int lane = tid & 31;
  const int c = lane & 15;               // N-column / A row provider
  const int h = lane >> 4;               // lane half
  const int rowbase = (blockIdx.x * 8 + wave) * 16;

  v8f aK1a = zero8(), aK1b = zero8(), aP2 = zero8(), aK20 = zero8(), aK21 = zero8();

  // per-lane bases; inner loop advances by pure +8 floats (imm offsets)
  const float* Arow = hidden + (size_t)(rowbase + c) * CV_ + 4 * h;
  const float* w0 = WT + (size_t)(0 * 16 + c) * CV_ + 4 * h;
  const float* w1p = WT + (size_t)(1 * 16 + c) * CV_ + 4 * h;
  const float* w2p = WT + (size_t)(2 * 16 + c) * CV_ + 4 * h;
  const float* w3p = WT + (size_t)(3 * 16 + c) * CV_ + 4 * h;
  const float* w4p = WT + (size_t)(4 * 16 + c) * CV_ + 4 * h;

#pragma unroll 4
  for (int kp = 0; kp < CV_ / 8; ++kp) {
    __builtin_prefetch(Arow + kp * 8 + 256, 0, 0);
    v4f a  = *(const v4f*)(Arow + kp * 8);
    v4f b0 = *(const v4f*)(w0  + kp * 8);
    v4f b1 = *(const v4f*)(w1p + kp * 8);
    v4f b2 = *(const v4f*)(w2p + kp * 8);
    v4f b3 = *(const v4f*)(w3p + kp * 8);
    v4f b4 = *(const v4f*)(w4p + kp * 8);
    aK1a = wmma_f32(lo2(a), lo2(b0), aK1a);
    aK1b = wmma_f32(lo2(a), lo2(b1), aK1b);
    aP2  = wmma_f32(lo2(a), lo2(b2), aP2);
    aK20 = wmma_f32(lo2(a), lo2(b3), aK20);
    aK21 = wmma_f32(lo2(a), lo2(b4), aK21);
    aK1a = wmma_f32(hi2(a), hi2(b0), aK1a);
    aK1b = wmma_f32(hi2(a), hi2(b1), aK1b);
    aP2  = wmma_f32(hi2(a), hi2(b2), aP2);
    aK20 = wmma_f32(hi2(a), hi2(b3), aK20);
    aK21 = wmma_f32(hi2(a), hi2(b4), aK21);
  }

  // biases + emit. C layout: VGPR j -> (row = j + 8h, col = c)
  const float bA = bk1[c], bBv = bk1[16 + c], bQ = bq2[c];
  const float b20 = bk2[c], b21 = bk2[CM2_ + c];
#pragma unroll
  for (int j = 0; j < 8; ++j) {
    const int row = j + 8 * h;
    lds_k1[wave][row * 32 + c]      = aK1a[j] + bA;
    lds_k1[wave][row * 32 + 16 + c] = aK1b[j] + bBv;
    p2o[(size_t)(rowbase + row) * CM2_ + c] = aP2[j] + bQ;
    // grouped k2: nv group == row parity
    k2o[(size_t)(rowbase + row) * CM2_ + c] =
        ((row & 1) == 0) ? (aK20[j] + b20) : (aK21[j] + b21);
  }
  __syncthreads();

  // ---- Phase 2: att1 over 512 rules ----
  // Preload k1 tile as A operand with the same chunk-pair mapping.
  v4f a1[4];
#pragma unroll
  for (int kp = 0; kp < 4; ++kp)
    a1[kp] = *(const v4f*)&lds_k1[wave][c * 32 + kp * 8 + 4 * h];

  float bestv[8]; int besti[8];
#pragma unroll
  for (int j = 0; j < 8; ++j) { bestv[j] = -3.402823466e38f; besti[j] = 0; }

  for (int rt = 0; rt < NR_ / 16; ++rt) {
    // B tile: element (K=m, N=r) = r1[r][m]; lane owns rule r = rt*16 + c
    const float* rb = r1w + (size_t)(rt * 16 + c) * CM1_ + 4 * h;
    v8f acc = zero8();
#pragma unroll
    for (int kp = 0; kp < 4; ++kp) {
      v4f bb = *(const v4f*)(rb + kp * 8);
      acc = wmma_f32(lo2(a1[kp]), lo2(bb), acc);
      acc = wmma_f32(hi2(a1[kp]), hi2(bb), acc);
    }
    const int r = rt * 16 + c;   // strictly increasing per lane
#pragma unroll
    for (int j = 0; j < 8; ++j) {
      float v = acc[j];
      if (v > bestv[j]) { bestv[j] = v; besti[j] = r; }
    }
  }

  // cross-lane reduce per row (16 rule-columns per row across half-wave lanes)
#pragma unroll
  for (int j = 0; j < 8; ++j) {
    const int row = j + 8 * h;
    lds_rv[wave][row * 16 + c] = bestv[j];
    lds_ri[wave][row * 16 + c] = besti[j];
  }
  __syncthreads();
  if (lane < 16) {
    const int row = lane;
    float bv = lds_rv[wave][row * 16];
    int   bi = lds_ri[wave][row * 16];
#pragma unroll
    for (int cc = 1; cc < 16; ++cc) {
      float v = lds_rv[wave][row * 16 + cc];
      int   i = lds_ri[wave][row * 16 + cc];
      if (v > bv || (v == bv && i < bi)) { bv = v; bi = i; }
    }
    bestvo[rowbase + row] = bv;
    bestio[rowbase + row] = bi;
  }
}

// ---------------------------------------------------------------------------
// Kernel C: per-batch finalize. Selection, att2, 4->32->2 MLP for the
// selected rule only (reference computes all 512 rules then discards).
// Positive scale factors 1/sqrt(CM) are irrelevant to argmax.
// ---------------------------------------------------------------------------
__global__ __launch_bounds__(256) void finalize_kernel(
    const float* __restrict__ hidden,
    const float* __restrict__ p2w, const float* __restrict__ k2w,
    const float* __restrict__ bestv, const int* __restrict__ besti,
    const float* __restrict__ w1, const float* __restrict__ b1,
    const float* __restrict__ w2, const float* __restrict__ b2,
    float* __restrict__ out) {
  const int b = blockIdx.x * 256 + threadIdx.x;
  if (b >= B_) return;
  const float v0 = bestv[2 * b],  v1 = bestv[2 * b + 1];
  const int   r0 = besti[2 * b],  r1i = besti[2 * b + 1];
  // flat tie-break: idx = r*2 + n ; (v1==v0) -> pick smaller r (then n=0)
  int nstar, rstar;
  if (v1 > v0 || (v1 == v0 && r1i < r0)) { nstar = 1; rstar = r1i; }
  else                                   { nstar = 0; rstar = r0;  }

  const float* q2 = p2w + (size_t)(2 * b + nstar) * CM2_;
  float a20 = 0.f, a21 = 0.f;
#pragma unroll
  for (int m = 0; m < CM2_; ++m) {
    a20 += q2[m] * k2w[(size_t)(2 * b) * CM2_ + m];
    a21 += q2[m] * k2w[(size_t)(2 * b + 1) * CM2_ + m];
  }
  const int n2 = (a21 > a20) ? 1 : 0;   // first-occurrence argmax

  const float d0 = hidden[(size_t)(2 * b + nstar) * CV_ + 0];
  const float d1 = hidden[(size_t)(2 * b + nstar) * CV_ + 1];
  const float d2 = hidden[(size_t)(2 * b + n2) * CV_ + 0];
  const float d3 = hidden[(size_t)(2 * b + n2) * CV_ + 1];

  const float* W1 = w1 + (size_t)rstar * 4 * 32;   // [d][e]
  const float* B1 = b1 + (size_t)rstar * 32;
  const float* W2 = w2 + (size_t)rstar * 32 * 2;   // [e][o]
  const float* B2 = b2 + (size_t)rstar * 2;
  float y0 = B2[0], y1 = B2[1];
#pragma unroll 8
  for (int e = 0; e < 32; ++e) {
    float hv = d0 * W1[e] + d1 * W1[32 + e] + d2 * W1[64 + e] + d3 * W1[96 + e] + B1[e];
    hv = fmaxf(hv, 0.f);
    y0 += hv * W2[e * 2];
    y1 += hv * W2[e * 2 + 1];
  }
  // out: (B,2,2) then mask_p: (B,2), concatenated flat
  out[b * 4 + 0] = (nstar == 0) ? y0 : 0.f;
  out[b * 4 + 1] = (nstar == 0) ? y1 : 0.f;
  out[b * 4 + 2] = (nstar == 1) ? y0 : 0.f;
  out[b * 4 + 3] = (nstar == 1) ? y1 : 0.f;
  float* mp = out + (size_t)B_ * NV_ * 2;
  mp[b * 2 + 0] = (nstar == 0) ? 1.f : 0.f;
  mp[b * 2 + 1] = (nstar == 1) ? 1.f : 0.f;
}

// ---------------------------------------------------------------------------
extern "C" void kernel_launch(void* const* d_in, const int* in_sizes, int n_in,
                              void* d_out, int out_size, void* d_ws, size_t ws_size,
                              hipStream_t stream) {
  const float* hidden      = (const float*)d_in[0];
  const float* rule_embeds = (const float*)d_in[1];
  const float* Wq1 = (const float*)d_in[2];
  const float* bq1 = (const float*)d_in[3];
  const float* Wk1 = (const float*)d_in[4];
  const float* bk1 = (const float*)d_in[5];
  const float* Wq2 = (const float*)d_in[6];
  const float* bq2 = (const float*)d_in[7];
  const float* Wk2 = (const float*)d_in[8];
  const float* bk2 = (const float*)d_in[9];
  const float* w1  = (const float*)d_in[10];
  const float* b1  = (const float*)d_in[11];
  const float* w2  = (const float*)d_in[12];
  const float* b2  = (const float*)d_in[13];

  // workspace carve (floats): r1 | WT | p2 | k2sel | bestv | besti
  float* ws  = (float*)d_ws;
  float* r1w = ws;                               // 512*32        = 16384
  float* WT  = r1w + NR_ * CM1_;                 // 5*16*1024     = 81920
  float* p2w = WT + NTILES_ * 16 * CV_;          // 16384*16      = 262144
  float* k2w = p2w + (size_t)ROWS_ * CM2_;       // 16384*16      = 262144
  float* bvw = k2w + (size_t)ROWS_ * CM2_;       // 16384
  int*   biw = (int*)(bvw + ROWS_);              // 16384

  r1_kernel<<<(NR_ * CM1_) / 256, 256, 0, stream>>>(rule_embeds, Wq1, bq1, r1w);
  wt_kernel<<<(NTILES_ * 16 * CV_) / 256, 256, 0, stream>>>(Wk1, Wq2, Wk2, WT);
  proj_att1_kernel<<<ROWS_ / (8 * 16), 256, 0, stream>>>(
      hidden, WT, bk1, bq2, bk2, r1w, p2w, k2w, bvw, biw);
  finalize_kernel<<<B_ / 256, 256, 0, stream>>>(
      hidden, p2w, k2w, bvw, biw, w1, b1, w2, b2, (float*)d_out);
}